// crf_18459769438644
// MI455X (gfx1250) — compile-verified
//
#include <hip/hip_runtime.h>
#include <hip/hip_bf16.h>

// CRF forward (partition function) on gfx1250.
// Per step, logsumexp recurrence is rewritten as an f32 GEMM:
//   new[i,b] = h_t[i,b] + M[b] + log( sum_j exp(trans[i,j]) * exp(score[j,b]-M[b]) )
// and the 64x16x64 product runs on V_WMMA_F32_16X16X4_F32 (16 chained WMMAs,
// 4 independent accumulator chains). A-fragments (exp(trans)) are resident in
// VGPRs for the whole T=1024 scan; p and h_t are staged in padded LDS with
// double-buffered prefetch of h_{t+1}.

typedef __attribute__((ext_vector_type(2))) float v2f;
typedef __attribute__((ext_vector_type(8))) float v8f;

#define NTAGS   64
#define SOS_IDX 1
#define EOS_IDX 2
#define NEGV    (-10000.0f)
#define BT      16      // batch tile per workgroup
#define NW      4       // waves per workgroup (one 16-row tag slice each)

__global__ __launch_bounds__(128)
void crf_fwd_kernel(const float* __restrict__ h,
                    const float* __restrict__ mask,
                    const float* __restrict__ trans,
                    float* __restrict__ out,
                    int T)
{
    __shared__ float sc[NTAGS][17];      // scores, padded (bank-conflict free cols)
    __shared__ float pb[NTAGS][17];      // p = exp(score - M)
    __shared__ float hb[2][BT][68];      // h_t staging, double buffered, padded
    __shared__ float Mred[NW][2][BT];    // per-wave-half column maxima
    __shared__ float MC[BT];             // M[b] broadcast for current step
    __shared__ float lenP[8][BT];        // partial mask sums

    const int tid  = threadIdx.x;
    const int lane = tid & 31;
    const int w    = tid >> 5;           // wave id == tag-row tile (rows 16w..16w+15)
    const int n    = tid & 15;           // batch column within tile
    const int half = lane >> 4;
    const int b0   = blockIdx.x * BT;

    // ---------------- init ----------------
    {   // initial scores: NEG everywhere, 0 at SOS
        int jb = (tid >> 4) * 8;
        for (int r = 0; r < 8; ++r) {
            int j = jb + r;
            sc[j][n] = (j == SOS_IDX) ? 0.0f : NEGV;
        }
    }
    if (tid < NW * 2 * BT) ((float*)Mred)[tid] = 0.0f;   // max of init scores = 0

    {   // per-batch sequence length from mask (partial sums, no atomics)
        int t0 = w * (T / 4) + half * (T / 8);
        const float* mrow = mask + (size_t)(b0 + n) * T;
        float s = 0.0f;
        for (int k = 0; k < T / 8; ++k) s += mrow[t0 + k];
        lenP[w * 2 + half][n] = s;
    }

    // A fragments: E = exp(trans), 16x4 f32 layout per chunk
    // lane<16: row=lane, K = 4c+{0,1}; lane>=16: row=lane-16, K = 4c+{2,3}
    v2f afr[16];
    {
        const float* trow = trans + (size_t)(w * 16 + n) * NTAGS;
        #pragma unroll
        for (int c = 0; c < 16; ++c) {
            int k = 4 * c + half * 2;
            afr[c].x = __expf(trow[k]);
            afr[c].y = __expf(trow[k + 1]);
        }
    }

    // stage h_0 (coalesced float4 loads)
    for (int q = tid; q < BT * 16; q += 128) {
        int bb = q >> 4, fi = (q & 15) * 4;
        float4 v = *(const float4*)(h + ((size_t)(b0 + bb) * T) * NTAGS + fi);
        *(float4*)&hb[0][bb][fi] = v;
    }
    __syncthreads();

    int lenN;
    {
        float s = 0.0f;
        #pragma unroll
        for (int k = 0; k < 8; ++k) s += lenP[k][n];
        lenN = (int)(s + 0.5f);
    }

    // ---------------- sequential scan over T ----------------
    for (int t = 0; t < T; ++t) {
        const int cur = t & 1, nxt = cur ^ 1;

        // prefetch h_{t+1} to registers (hidden behind Phase A)
        float4 pf0, pf1;
        const bool do_pf = (t + 1 < T);
        const int q0 = tid, q1 = tid + 128;
        const int bb0 = q0 >> 4, fi0 = (q0 & 15) * 4;
        const int bb1 = q1 >> 4, fi1 = (q1 & 15) * 4;
        if (do_pf) {
            pf0 = *(const float4*)(h + ((size_t)(b0 + bb0) * T + (t + 1)) * NTAGS + fi0);
            pf1 = *(const float4*)(h + ((size_t)(b0 + bb1) * T + (t + 1)) * NTAGS + fi1);
        }

        // ---- Phase A: M[b] and p = exp(score - M) ----
        float Mcur;
        {
            float m0 = ((float*)Mred)[n];
            #pragma unroll
            for (int k = 1; k < 8; ++k) m0 = fmaxf(m0, ((float*)Mred)[k * 16 + n]);
            Mcur = m0;
        }
        {
            int jb = (tid >> 4) * 8;
            #pragma unroll
            for (int r = 0; r < 8; ++r) {
                int j = jb + r;
                pb[j][n] = __expf(sc[j][n] - Mcur);
            }
        }
        if (tid < BT) MC[tid] = Mcur;
        if (do_pf) {
            *(float4*)&hb[nxt][bb0][fi0] = pf0;
            *(float4*)&hb[nxt][bb1][fi1] = pf1;
        }
        __syncthreads();

        // ---- Phase B: S = E * p via WMMA f32 16x16x4, then log + mask ----
        v8f acc[4];
        #pragma unroll
        for (int q = 0; q < 4; ++q)
            acc[q] = (v8f){0.f, 0.f, 0.f, 0.f, 0.f, 0.f, 0.f, 0.f};

        #pragma unroll
        for (int c = 0; c < 16; ++c) {
            int k = 4 * c + half * 2;     // B 4x16: lanes<16 K={0,1}, lanes>=16 K={2,3}
            v2f bf;
            bf.x = pb[k][n];
            bf.y = pb[k + 1][n];
            acc[c >> 2] = __builtin_amdgcn_wmma_f32_16x16x4_f32(
                false, afr[c], false, bf, (short)0, acc[c >> 2], false, false);
        }
        v8f S = acc[0] + acc[1] + acc[2] + acc[3];

        const float Mn = MC[n];
        const bool  live = (t < lenN);
        float colmax = -3.0e38f;
        #pragma unroll
        for (int r = 0; r < 8; ++r) {
            int il = r + 8 * half;        // C/D layout: lanes<16 M=r, lanes>=16 M=r+8
            int i  = w * 16 + il;
            float hv  = hb[cur][n][i];
            float nv  = hv + Mn + __logf(S[r]);
            float ov  = sc[i][n];
            float sel = live ? nv : ov;
            sc[i][n]  = sel;
            colmax = fmaxf(colmax, sel);
        }
        Mred[w][half][n] = colmax;
        __syncthreads();
    }

    // ---------------- final: out[b] = LSE_j(score[j,b] + trans[EOS,j]) ----------------
    if (tid < BT) {
        const float* te = trans + EOS_IDX * NTAGS;
        float mx = -3.0e38f;
        for (int j = 0; j < NTAGS; ++j) mx = fmaxf(mx, sc[j][tid] + te[j]);
        float s = 0.0f;
        for (int j = 0; j < NTAGS; ++j) s += __expf(sc[j][tid] + te[j] - mx);
        out[b0 + tid] = mx + __logf(s);
    }
}

extern "C" void kernel_launch(void* const* d_in, const int* in_sizes, int n_in,
                              void* d_out, int out_size, void* d_ws, size_t ws_size,
                              hipStream_t stream) {
    (void)n_in; (void)d_ws; (void)ws_size;
    const float* h     = (const float*)d_in[0];
    const float* mask  = (const float*)d_in[1];
    const float* trans = (const float*)d_in[2];
    float* out = (float*)d_out;

    const int B = out_size;               // 256
    const int T = in_sizes[1] / B;        // 1024
    dim3 grid(B / BT), block(128);
    hipLaunchKernelGGL(crf_fwd_kernel, grid, block, 0, stream, h, mask, trans, out, T);
}